// SS2DBlock_32899449488068
// MI455X (gfx1250) — compile-verified
//
#include <hip/hip_runtime.h>
#include <hip/hip_bf16.h>

typedef float v2f __attribute__((ext_vector_type(2)));
typedef float v8f __attribute__((ext_vector_type(8)));

#define DIMC   96
#define DSTATE 16
#define DIN    192
#define DTR    6
#define KDIR   4
#define BATCH  2
#define HH     56
#define WW     56
#define LL     (HH*WW)          // 3136
#define XPC    (DTR + 2*DSTATE) // 38

// ---------------------------------------------------------------------------
// fp32 WMMA GEMM: C[M,N] = A[M,K] * B[K,N], row-major.
// One wave computes a 16(M) x 32(N) tile: two accumulators sharing one A
// fragment, K stepped by 4 via V_WMMA_F32_16X16X4_F32.
// Full-M tiles take a wave-uniform unguarded path (b64 A loads, pointer-
// increment addressing); the single partial-M tile (x-proj, M=38) uses
// clamped addresses + multiply-by-mask so there is no exec-mask churn.
// ---------------------------------------------------------------------------
__global__ __launch_bounds__(32)
void gemm_f32_wmma(const float* __restrict__ Amat, const float* __restrict__ Bmat,
                   float* __restrict__ Cmat, int M, int K, int N,
                   int aMod, long aStride, long bStride, long cStride) {
    const int z = blockIdx.z;
    Amat += (long)(z % aMod) * aStride;
    Bmat += (long)z * bStride;
    Cmat += (long)z * cStride;

    const int lane  = threadIdx.x & 31;
    const int row   = lane & 15;        // M index for A-frag, N index for B-frag
    const int khalf = (lane >> 4) * 2;  // K pair base (0 or 2)
    const int m0    = blockIdx.y * 16;
    const int n0    = blockIdx.x * 32;

    v8f acc0 = {}, acc1 = {};

    // B fragment pointers: row khalf (x) and khalf+1 (y); cols n0+row / +16.
    const float* b0 = Bmat + (long)khalf * N + n0 + row;
    const float* b1 = b0 + N;
    const long   bInc = 4L * N;

    if (m0 + 16 <= M) {
        // ---- fast path: full 16-row tile, unguarded b64 A loads ----
        const float* ap = Amat + (long)(m0 + row) * K + khalf;
#pragma unroll 2
        for (int kk = 0; kk < K; kk += 4) {
            const v2f a = *(const v2f*)ap;          // A[m][kk+khalf], +1
            v2f bl, bh;
            bl.x = b0[0];  bh.x = b0[16];
            bl.y = b1[0];  bh.y = b1[16];
            acc0 = __builtin_amdgcn_wmma_f32_16x16x4_f32(
                       false, a, false, bl, (short)0, acc0, false, false);
            acc1 = __builtin_amdgcn_wmma_f32_16x16x4_f32(
                       false, a, false, bh, (short)0, acc1, false, false);
            ap += 4;  b0 += bInc;  b1 += bInc;
        }
    } else {
        // ---- partial tile: clamped address + mask, no divergence ----
        const int   am   = m0 + row;
        const float mask = (am < M) ? 1.f : 0.f;
        const int   amc  = (am < M) ? am : (M - 1);
        const float* ap = Amat + (long)amc * K + khalf;
#pragma unroll 2
        for (int kk = 0; kk < K; kk += 4) {
            v2f a;
            a.x = ap[0] * mask;
            a.y = ap[1] * mask;
            v2f bl, bh;
            bl.x = b0[0];  bh.x = b0[16];
            bl.y = b1[0];  bh.y = b1[16];
            acc0 = __builtin_amdgcn_wmma_f32_16x16x4_f32(
                       false, a, false, bl, (short)0, acc0, false, false);
            acc1 = __builtin_amdgcn_wmma_f32_16x16x4_f32(
                       false, a, false, bh, (short)0, acc1, false, false);
            ap += 4;  b0 += bInc;  b1 += bInc;
        }
    }

    // C/D 16x16: VGPR r -> M = r + 8*(lane/16), lane&15 -> N
    const int half = lane >> 4;
    float* cp = Cmat + n0 + (lane & 15);
#pragma unroll
    for (int r = 0; r < 8; ++r) {
        const int m = m0 + r + 8 * half;
        if (m < M) {
            cp[(long)m * N]      = acc0[r];
            cp[(long)m * N + 16] = acc1[r];
        }
    }
}

// ---------------------------------------------------------------------------
// Depthwise 3x3 SAME conv (+bias) on xc = xz[:, :192] followed by SiLU.
// ---------------------------------------------------------------------------
__global__ void conv_silu_kernel(const float* __restrict__ xz,
                                 const float* __restrict__ w_conv,
                                 const float* __restrict__ b_conv,
                                 float* __restrict__ xc_act) {
    const int idx = blockIdx.x * blockDim.x + threadIdx.x;
    if (idx >= BATCH * DIN * LL) return;
    const int l = idx % LL;
    const int t = idx / LL;
    const int d = t % DIN;
    const int b = t / DIN;
    const int hh = l / WW, ww = l % WW;

    float s = b_conv[d];
    const float* src = xz + (long)(b * 2 * DIN + d) * LL;
#pragma unroll
    for (int dy = -1; dy <= 1; ++dy) {
        const int y = hh + dy;
        if (y < 0 || y >= HH) continue;
#pragma unroll
        for (int dx = -1; dx <= 1; ++dx) {
            const int x = ww + dx;
            if (x < 0 || x >= WW) continue;
            s += w_conv[d * 9 + (dy + 1) * 3 + (dx + 1)] * src[y * WW + x];
        }
    }
    const float act = s / (1.f + __expf(-s));   // SiLU
    xc_act[(long)(b * DIN + d) * LL + l] = act;
}

// ---------------------------------------------------------------------------
// Build the 4 scan orderings: k0 row-major, k1 transposed, k2/k3 flipped.
// ---------------------------------------------------------------------------
__global__ void build_xs_kernel(const float* __restrict__ xc_act,
                                float* __restrict__ xs) {
    const int idx = blockIdx.x * blockDim.x + threadIdx.x;
    if (idx >= BATCH * DIN * LL) return;
    const int l = idx % LL;
    const int t = idx / LL;
    const int d = t % DIN;
    const int b = t / DIN;
    const int hh = l / WW, ww = l % WW;
    const int lt = ww * HH + hh;

    const float v = xc_act[(long)(b * DIN + d) * LL + l];
    const long  S = (long)DIN * LL;                       // per-k stride
    float* base = xs + ((long)(b * KDIR) * DIN + d) * LL; // k = 0 base
    base[l]                     = v;   // k0
    base[S + lt]                = v;   // k1
    base[2 * S + (LL - 1 - l)]  = v;   // k2
    base[3 * S + (LL - 1 - lt)] = v;   // k3
}

// ---------------------------------------------------------------------------
// delta = softplus(w_dt @ dts + b_dt)  (DT_RANK = 6 contraction, scalar)
// ---------------------------------------------------------------------------
__global__ void delta_kernel(const float* __restrict__ x_dbl,
                             const float* __restrict__ w_dt,
                             const float* __restrict__ b_dt,
                             float* __restrict__ delta) {
    const int idx = blockIdx.x * blockDim.x + threadIdx.x;
    if (idx >= BATCH * KDIR * DIN * LL) return;
    const int l  = idx % LL;
    const int t  = idx / LL;
    const int d  = t % DIN;
    const int bk = t / DIN;
    const int k  = bk & 3;

    float ac = b_dt[k * DIN + d];
#pragma unroll
    for (int r = 0; r < DTR; ++r)
        ac += w_dt[(k * DIN + d) * DTR + r] * x_dbl[((long)bk * XPC + r) * LL + l];
    const float sp = (ac > 20.f) ? ac : log1pf(__expf(ac));
    delta[(long)(bk * DIN + d) * LL + l] = sp;
}

// ---------------------------------------------------------------------------
// Selective scan: one block per (b,k), one thread per d channel.
// 16-state recurrence in registers; per-l B/C vectors staged through LDS in
// 16-step chunks (shared by all 192 channels of the block).
// ---------------------------------------------------------------------------
__global__ __launch_bounds__(DIN)
void scan_kernel(const float* __restrict__ delta, const float* __restrict__ xs,
                 const float* __restrict__ x_dbl, const float* __restrict__ A_log,
                 const float* __restrict__ D_skip, float* __restrict__ ys) {
    const int bk = blockIdx.x;   // b*4 + k
    const int k  = bk & 3;
    const int d  = threadIdx.x;  // 0..191

    float hreg[DSTATE], Areg[DSTATE];
#pragma unroll
    for (int n = 0; n < DSTATE; ++n) {
        hreg[n] = 0.f;
        Areg[n] = -__expf(A_log[(k * DIN + d) * DSTATE + n]);
    }
    const float dskip = D_skip[k * DIN + d];

    const float* dptr  = delta + (long)(bk * DIN + d) * LL;
    const float* xptr  = xs    + (long)(bk * DIN + d) * LL;
    const float* Bbase = x_dbl + ((long)bk * XPC + DTR) * LL;
    const float* Cbase = x_dbl + ((long)bk * XPC + DTR + DSTATE) * LL;
    float*       yptr  = ys    + (long)(bk * DIN + d) * LL;

    __shared__ float sB[16][DSTATE];
    __shared__ float sC[16][DSTATE];

    for (int l0 = 0; l0 < LL; l0 += 16) {
        // cooperative stage of B/C for the next 16 timesteps
        for (int idx = threadIdx.x; idx < 16 * 32; idx += DIN) {
            const int j = idx >> 5, c = idx & 31;
            if (c < DSTATE) sB[j][c]          = Bbase[(long)c * LL + l0 + j];
            else            sC[j][c - DSTATE] = Cbase[(long)(c - DSTATE) * LL + l0 + j];
        }
        __syncthreads();
        if (l0 + 80 < LL) {
            __builtin_prefetch(dptr + l0 + 80, 0, 1);
            __builtin_prefetch(xptr + l0 + 80, 0, 1);
        }
#pragma unroll 4
        for (int j = 0; j < 16; ++j) {
            const int l = l0 + j;
            const float dv = dptr[l];
            const float xv = xptr[l];
            const float dx = dv * xv;
            float y = 0.f;
#pragma unroll
            for (int n = 0; n < DSTATE; ++n) {
                hreg[n] = __expf(dv * Areg[n]) * hreg[n] + dx * sB[j][n];
                y += hreg[n] * sC[j][n];
            }
            yptr[l] = y + dskip * xv;
        }
        __syncthreads();
    }
}

// ---------------------------------------------------------------------------
// Combine 4 scan directions, LayerNorm over d (wave32 per pixel), gate with
// SiLU(z). Produces yact[b,d,l] for the output GEMM.
// ---------------------------------------------------------------------------
__global__ __launch_bounds__(128)
void combine_ln_gate_kernel(const float* __restrict__ ys,
                            const float* __restrict__ xz,
                            const float* __restrict__ ln_g,
                            const float* __restrict__ ln_b,
                            float* __restrict__ yact) {
    const int wave = (blockIdx.x * (blockDim.x >> 5)) + (threadIdx.x >> 5);
    const int lane = threadIdx.x & 31;
    if (wave >= BATCH * LL) return;
    const int b = wave / LL;
    const int l = wave % LL;
    const int hh = l / WW, ww = l % WW;
    const int lt = ww * HH + hh;

    const long S = (long)DIN * LL;                        // per-k stride
    const float* base = ys + (long)(b * KDIR) * DIN * LL; // k = 0

    float v[DIN / 32];
    float s1 = 0.f, s2 = 0.f;
#pragma unroll
    for (int i = 0; i < DIN / 32; ++i) {
        const int d = lane + 32 * i;
        const long rowo = (long)d * LL;
        const float y0 = base[rowo + l];
        const float y2 = base[2 * S + rowo + (LL - 1 - l)];
        const float y1 = base[S + rowo + lt];
        const float y3 = base[3 * S + rowo + (LL - 1 - lt)];
        const float vv = y0 + y2 + y1 + y3;
        v[i] = vv;
        s1 += vv;
        s2 += vv * vv;
    }
#pragma unroll
    for (int off = 16; off > 0; off >>= 1) {
        s1 += __shfl_xor(s1, off, 32);
        s2 += __shfl_xor(s2, off, 32);
    }
    const float mu   = s1 * (1.f / DIN);
    const float var  = s2 * (1.f / DIN) - mu * mu;
    const float rstd = rsqrtf(var + 1e-5f);

#pragma unroll
    for (int i = 0; i < DIN / 32; ++i) {
        const int d = lane + 32 * i;
        const float yn = (v[i] - mu) * rstd * ln_g[d] + ln_b[d];
        const float zv = xz[(long)(b * 2 * DIN + DIN + d) * LL + l];
        const float gate = zv / (1.f + __expf(-zv));      // SiLU(z)
        yact[(long)(b * DIN + d) * LL + l] = yn * gate;
    }
}

// ---------------------------------------------------------------------------
extern "C" void kernel_launch(void* const* d_in, const int* in_sizes, int n_in,
                              void* d_out, int out_size, void* d_ws, size_t ws_size,
                              hipStream_t stream) {
    const float* x      = (const float*)d_in[0];   // (2,96,56,56)
    const float* w_in   = (const float*)d_in[1];   // (384,96)
    const float* w_conv = (const float*)d_in[2];   // (192,1,3,3)
    const float* b_conv = (const float*)d_in[3];   // (192,)
    const float* w_xprj = (const float*)d_in[4];   // (4,38,192)
    const float* w_dt   = (const float*)d_in[5];   // (4,192,6)
    const float* b_dt   = (const float*)d_in[6];   // (4,192)
    const float* A_log  = (const float*)d_in[7];   // (4,192,16)
    const float* D_skip = (const float*)d_in[8];   // (4,192)
    const float* ln_g   = (const float*)d_in[9];   // (192,)
    const float* ln_b   = (const float*)d_in[10];  // (192,)
    const float* w_out  = (const float*)d_in[11];  // (96,192)
    float* out = (float*)d_out;                    // (2,96,56,56)

    // workspace layout (floats)
    float* ws = (float*)d_ws;
    float* xz     = ws;                                   // 2*384*L
    float* xc_act = xz     + (long)BATCH * 2 * DIN * LL;  // 2*192*L
    float* xs     = xc_act + (long)BATCH * DIN * LL;      // 2*4*192*L
    float* x_dbl  = xs     + (long)BATCH * KDIR * DIN * LL;      // 2*4*38*L
    float* delta  = x_dbl  + (long)BATCH * KDIR * XPC * LL;      // 2*4*192*L
    float* ys     = delta  + (long)BATCH * KDIR * DIN * LL;      // 2*4*192*L
    float* yact   = ys     + (long)BATCH * KDIR * DIN * LL;      // 2*192*L

    // 1) input projection: xz[b] = w_in (384x96) * x[b] (96xL)
    gemm_f32_wmma<<<dim3(LL / 32, (2 * DIN) / 16, BATCH), 32, 0, stream>>>(
        w_in, x, xz, 2 * DIN, DIMC, LL,
        1, 0L, (long)DIMC * LL, (long)2 * DIN * LL);

    // 2) depthwise conv + SiLU
    {
        const int n = BATCH * DIN * LL;
        conv_silu_kernel<<<(n + 255) / 256, 256, 0, stream>>>(xz, w_conv, b_conv, xc_act);
    }

    // 3) build 4 scan orderings
    {
        const int n = BATCH * DIN * LL;
        build_xs_kernel<<<(n + 255) / 256, 256, 0, stream>>>(xc_act, xs);
    }

    // 4) x_dbl[b,k] = w_xproj[k] (38x192) * xs[b,k] (192xL)
    gemm_f32_wmma<<<dim3(LL / 32, (XPC + 15) / 16, BATCH * KDIR), 32, 0, stream>>>(
        w_xprj, xs, x_dbl, XPC, DIN, LL,
        KDIR, (long)XPC * DIN, (long)DIN * LL, (long)XPC * LL);

    // 5) delta = softplus(w_dt @ dts + b_dt)
    {
        const int n = BATCH * KDIR * DIN * LL;
        delta_kernel<<<(n + 255) / 256, 256, 0, stream>>>(x_dbl, w_dt, b_dt, delta);
    }

    // 6) selective scan
    scan_kernel<<<BATCH * KDIR, DIN, 0, stream>>>(delta, xs, x_dbl, A_log, D_skip, ys);

    // 7) combine directions + LayerNorm + SiLU gating
    {
        const int waves = BATCH * LL;                  // one wave32 per pixel
        combine_ln_gate_kernel<<<(waves + 3) / 4, 128, 0, stream>>>(ys, xz, ln_g, ln_b, yact);
    }

    // 8) output projection: out[b] = w_out (96x192) * yact[b] (192xL)
    gemm_f32_wmma<<<dim3(LL / 32, DIMC / 16, BATCH), 32, 0, stream>>>(
        w_out, yact, out, DIMC, DIN, LL,
        1, 0L, (long)DIN * LL, (long)DIMC * LL);
}